// RepresentationsFusing_18098992185824
// MI455X (gfx1250) — compile-verified
//
#include <hip/hip_runtime.h>
#include <cmath>

#define B_    8
#define S_    50
#define I_    20
#define C_    64
#define N_    4096
#define MMAX_ 1024   // S_*I_ = 1000 distinct items max per batch

typedef float v2f __attribute__((ext_vector_type(2)));
typedef float v8f __attribute__((ext_vector_type(8)));

// LDS chunk layout: 16 rows x 64 floats, row stride padded to 68 dwords (272 B,
// 16B-aligned) => ds_load_b64 banks = 4r + 4kk + 2h cover all 64 banks once.
#define LROW_ 68
#define LCHUNK_ (16 * LROW_)          // floats per chunk buffer (1088)
#define LCHUNKB_ (LCHUNK_ * 4)        // bytes (4352)

// ---------------------------------------------------------------------------
// 1) Per-(batch,item) occurrence counts. Integer atomics => deterministic.
// ---------------------------------------------------------------------------
__global__ void count_k(const int* __restrict__ ids, const int* __restrict__ lens,
                        int* __restrict__ icounts) {
  int rIdx = blockIdx.x * blockDim.x + threadIdx.x;
  if (rIdx >= B_ * S_ * I_) return;
  int b = rIdx / (S_ * I_);
  int s = (rIdx % (S_ * I_)) / I_;
  if (s >= lens[b]) return;
  atomicAdd(&icounts[b * N_ + ids[rIdx]], 1);
}

// ---------------------------------------------------------------------------
// 2) Deterministic compaction: exclusive scan of present flags per batch.
// ---------------------------------------------------------------------------
__global__ void compact_k(const int* __restrict__ icounts, int* __restrict__ midx,
                          int* __restrict__ Mb) {
  __shared__ int part[256];
  int b = blockIdx.x, t = threadIdx.x;
  const int PER = N_ / 256;   // 16
  int m0 = t * PER;
  int cnt = 0;
  for (int j = 0; j < PER; ++j)
    cnt += (icounts[b * N_ + m0 + j] > 0) ? 1 : 0;
  part[t] = cnt;
  __syncthreads();
  if (t == 0) {
    int acc = 0;
    for (int i = 0; i < 256; ++i) { int v = part[i]; part[i] = acc; acc += v; }
    Mb[b] = acc;
  }
  __syncthreads();
  int slot = part[t];
  for (int j = 0; j < PER; ++j) {
    int m = m0 + j;
    if (icounts[b * N_ + m] > 0) midx[b * MMAX_ + slot++] = m;
  }
}

// ---------------------------------------------------------------------------
// 3) Fill compacted mean rows; fixed-order scan => deterministic FP sums.
// ---------------------------------------------------------------------------
__global__ void fillmean_k(const float* __restrict__ inp, const int* __restrict__ ids,
                           const int* __restrict__ lens, const int* __restrict__ icounts,
                           const int* __restrict__ midx, const int* __restrict__ Mb,
                           float* __restrict__ meanc) {
  int blk = blockIdx.x;
  int b = blk / MMAX_;
  int j = blk % MMAX_;
  if (j >= Mb[b]) return;
  int m = midx[b * MMAX_ + j];
  int c = threadIdx.x;
  int len = lens[b];
  float acc = 0.0f;
  for (int s = 0; s < len; ++s) {
    const int*   idr = ids + (b * S_ + s) * I_;
    const float* inr = inp + (size_t)((b * S_ + s) * I_) * C_;
    for (int i = 0; i < I_; ++i)
      if (idr[i] == m) acc += inr[(size_t)i * C_ + c];
  }
  meanc[((size_t)(b * MMAX_ + j)) * C_ + c] = acc / (float)icounts[b * N_ + m];
}

// ---------------------------------------------------------------------------
// 4) g = emb @ proj_w via V_WMMA_F32_16X16X4_F32 (one wave per 16x16 tile).
// ---------------------------------------------------------------------------
__global__ void gemm_g_k(const float* __restrict__ emb, const float* __restrict__ pw,
                         float* __restrict__ g) {
  int tile = blockIdx.x;
  int nt = tile / (C_ / 16);
  int jt = tile % (C_ / 16);
  int lane = threadIdx.x;
  int r = lane & 15;
  int h = lane >> 4;
  int nrow = nt * 16 + r;
  int jcol = jt * 16 + r;
  v8f acc = {};
  #pragma unroll
  for (int k0 = 0; k0 < C_; k0 += 4) {
    int ka = k0 + 2 * h;
    v2f a, bm;
    a.x  = emb[nrow * C_ + ka];       a.y  = emb[nrow * C_ + ka + 1];
    bm.x = pw[ka * C_ + jcol];        bm.y = pw[(ka + 1) * C_ + jcol];
    acc = __builtin_amdgcn_wmma_f32_16x16x4_f32(false, a, false, bm,
                                                (short)0, acc, false, false);
  }
  #pragma unroll
  for (int i = 0; i < 8; ++i)
    g[(nt * 16 + 8 * h + i) * C_ + jt * 16 + r] = acc[i];
}

// ---------------------------------------------------------------------------
// 5) constn[n] = emb[n] . proj_b
// ---------------------------------------------------------------------------
__global__ void constn_k(const float* __restrict__ emb, const float* __restrict__ pb,
                         float* __restrict__ constn) {
  int n = blockIdx.x;
  int lane = threadIdx.x;
  float v = emb[n * C_ + 2 * lane] * pb[2 * lane]
          + emb[n * C_ + 2 * lane + 1] * pb[2 * lane + 1];
  #pragma unroll
  for (int off = 1; off < 32; off <<= 1) v += __shfl_xor(v, off);
  if (lane == 0) constn[n] = v;
}

// ---------------------------------------------------------------------------
// 6) Fused attention+projection, one wave per (b, 16-row n-tile).
//    mean chunks staged global->LDS with coalesced async b128 copies
//    (ASYNCcnt, double-buffered), WMMA f32 tiles fed from padded LDS.
// ---------------------------------------------------------------------------
__global__ void attn_k(const float* __restrict__ emb, const float* __restrict__ g,
                       const float* __restrict__ constn, const float* __restrict__ meanc,
                       const int* __restrict__ Mb, float* __restrict__ out) {
  __shared__ __align__(16) float lbuf[2 * LCHUNK_];   // double-buffered 2 x 4352 B

  int tile = blockIdx.x;
  int b  = tile / (N_ / 16);
  int nt = tile % (N_ / 16);
  int lane = threadIdx.x;
  int r = lane & 15;
  int h = lane >> 4;
  int nrow = nt * 16 + r;

  // Preload A operands for all 16 k-steps (emb rows and g rows).
  v2f Ae[16], Ag[16];
  #pragma unroll
  for (int kk = 0; kk < 16; ++kk) {
    int ka = kk * 4 + 2 * h;
    Ae[kk].x = emb[nrow * C_ + ka]; Ae[kk].y = emb[nrow * C_ + ka + 1];
    Ag[kk].x = g[nrow * C_ + ka];   Ag[kk].y = g[nrow * C_ + ka + 1];
  }

  int M = Mb[b];                       // >= 1 (batch_seq_length >= 1)
  int nch = (M + 15) >> 4;
  unsigned long long mcaddr =
      (unsigned long long)(uintptr_t)(meanc + (size_t)b * MMAX_ * C_);
  uint32_t lbase = (uint32_t)(uintptr_t)&lbuf[0];     // LDS byte offset

  // Coalesced async copy of one 16x64-f32 chunk: lane (h,r) moves 16 B of
  // row (2*it+h) at in-row offset r*16; tail rows clamped to M-1 (finite data;
  // their columns are masked to -inf later, and 0*finite == 0).
  auto issue_chunk = [&](int t) {
    #pragma unroll
    for (int it = 0; it < 8; ++it) {
      int row = 2 * it + h;
      int grow = t * 16 + row;
      grow = (grow < M) ? grow : (M - 1);
      unsigned long long ga = mcaddr + (unsigned long long)grow * (C_ * 4)
                              + (unsigned)(r * 16);
      uint32_t la = lbase + (uint32_t)((t & 1) * LCHUNKB_ + row * (LROW_ * 4) + r * 16);
      asm volatile("global_load_async_to_lds_b128 %0, %1, off"
                   :: "v"(la), "v"(ga) : "memory");
    }
  };

  issue_chunk(0);

  float rmax[8], rden[8], rnum[8];
  #pragma unroll
  for (int i = 0; i < 8; ++i) { rmax[i] = -INFINITY; rden[i] = 0.f; rnum[i] = 0.f; }

  for (int t = 0; t < nch; ++t) {
    int m0 = t << 4;
    if (t + 1 < nch) {
      // Buffer (t+1)&1 was last read one iteration ago; close the WAR window.
      asm volatile("s_wait_dscnt 0x0" ::: "memory");
      issue_chunk(t + 1);
      // In-order async completion: <=8 outstanding => chunk t has landed.
      asm volatile("s_wait_asynccnt 0x8" ::: "memory");
    } else {
      asm volatile("s_wait_asynccnt 0x0" ::: "memory");
    }

    // B operand from LDS: row r, k-pair ka = 4*kk + 2*h. Bank-conflict free.
    const float* lb = &lbuf[(t & 1) * LCHUNK_];
    v2f Bm[16];
    #pragma unroll
    for (int kk = 0; kk < 16; ++kk)
      Bm[kk] = *(const v2f*)&lb[r * LROW_ + kk * 4 + 2 * h];

    v8f sA = {};  // scores: emb . meanT
    v8f sQ = {};  // values: g . meanT
    #pragma unroll
    for (int kk = 0; kk < 16; ++kk) {
      sA = __builtin_amdgcn_wmma_f32_16x16x4_f32(false, Ae[kk], false, Bm[kk],
                                                 (short)0, sA, false, false);
      sQ = __builtin_amdgcn_wmma_f32_16x16x4_f32(false, Ag[kk], false, Bm[kk],
                                                 (short)0, sQ, false, false);
    }

    bool vmask = (m0 + r < M);
    // Online softmax. D element i on this lane: row M = 8*h + i, col m = m0 + r.
    #pragma unroll
    for (int i = 0; i < 8; ++i) {
      float s = sA[i];
      s = (s > 0.f) ? s : 0.2f * s;       // leaky_relu(0.2)
      if (!vmask) s = -INFINITY;          // mask absent/clamped items
      float q = sQ[i];
      float cmax = s;
      #pragma unroll
      for (int off = 1; off < 16; off <<= 1)
        cmax = fmaxf(cmax, __shfl_xor(cmax, off));
      float nmax = fmaxf(rmax[i], cmax);
      float p  = __expf(s - nmax);        // 0 for masked lanes
      float pq = p * q;
      #pragma unroll
      for (int off = 1; off < 16; off <<= 1) {
        p  += __shfl_xor(p,  off);
        pq += __shfl_xor(pq, off);
      }
      float scale = __expf(rmax[i] - nmax);   // exp(-inf)=0 on first chunk
      rden[i] = rden[i] * scale + p;
      rnum[i] = rnum[i] * scale + pq;
      rmax[i] = nmax;
    }
  }

  if (r == 0) {  // lane 0 -> rows 0..7, lane 16 -> rows 8..15
    #pragma unroll
    for (int i = 0; i < 8; ++i) {
      int n = nt * 16 + 8 * h + i;
      out[b * N_ + n] = rnum[i] / rden[i] + constn[n];
    }
  }
}

// ---------------------------------------------------------------------------
extern "C" void kernel_launch(void* const* d_in, const int* in_sizes, int n_in,
                              void* d_out, int out_size, void* d_ws, size_t ws_size,
                              hipStream_t stream) {
  const float* inp  = (const float*)d_in[0];   // (B,S,I,C)
  const float* emb  = (const float*)d_in[1];   // (N,C)
  const float* pw   = (const float*)d_in[2];   // (C,C)
  const float* pb   = (const float*)d_in[3];   // (C,)
  const int*   ids  = (const int*)d_in[4];     // (B,S,I)
  const int*   lens = (const int*)d_in[5];     // (B,)
  float* out = (float*)d_out;                  // (B,N)

  // Workspace layout (~3.3 MB total)
  int*   icounts = (int*)d_ws;                              // B*N
  float* meanc   = (float*)(icounts + (size_t)B_ * N_);     // B*MMAX*C
  float* g       = meanc + (size_t)B_ * MMAX_ * C_;         // N*C
  float* constn  = g + (size_t)N_ * C_;                     // N
  int*   midx    = (int*)(constn + N_);                     // B*MMAX
  int*   Mb      = midx + (size_t)B_ * MMAX_;               // B

  hipMemsetAsync(icounts, 0, (size_t)B_ * N_ * sizeof(int), stream);

  count_k   <<<(B_ * S_ * I_ + 255) / 256, 256, 0, stream>>>(ids, lens, icounts);
  compact_k <<<B_, 256, 0, stream>>>(icounts, midx, Mb);
  fillmean_k<<<B_ * MMAX_, C_, 0, stream>>>(inp, ids, lens, icounts, midx, Mb, meanc);
  gemm_g_k  <<<(N_ / 16) * (C_ / 16), 32, 0, stream>>>(emb, pw, g);
  constn_k  <<<N_, 32, 0, stream>>>(emb, pb, constn);
  attn_k    <<<B_ * (N_ / 16), 32, 0, stream>>>(emb, g, constn, meanc, Mb, out);
}